// SpatialAttention_77790447665639
// MI455X (gfx1250) — compile-verified
//
#include <hip/hip_runtime.h>

// SpatialAttention for MI455X (gfx1250, wave32).
// b=16, cl=512 channels, n=3136 pixels. HBM-bound (~0.5 FLOP/byte):
// floor = read x (102.8MB) + write out (102.8MB) ~ 9us @ 23.3 TB/s.
//
//  Pass 1 (WMMA): t[b,n] = theta.x[b,:,n], g[b,n] = gw.x[b,:,n] fused into one
//     chained V_WMMA_F32_16X16X4_F32 stream (A row0=theta, row1=gw, rows>=2 = 0;
//     full fp32, no precision loss). Each wave owns 32 pixels (two 16x16 D
//     accumulators sharing the A operand) -> 128B-contiguous row traffic.
//     t is wave-reduced + global_atomic_add_f32 into tsum[b]; g -> workspace.
//  Pass 2: float4 elementwise epilogue; x re-read is L2-hot (102.8MB < 192MB L2),
//     out uses non-temporal stores so the write stream doesn't evict x.

typedef __attribute__((ext_vector_type(2))) float v2f;
typedef __attribute__((ext_vector_type(4))) float v4f;
typedef __attribute__((ext_vector_type(8))) float v8f;

#define NB   16
#define CL   512
#define NPIX 3136
#define PAIRS (NPIX / 32)                    // 98 32-pixel tile-pairs per batch
#define WAVES_PER_BLOCK 2
#define BLOCKS_PER_B (PAIRS / WAVES_PER_BLOCK)  // 49

// ---------------------------------------------------------------------------
// Pass 1: fused theta/g GEMV via f32 WMMA (16x16x4), fp32 accumulate.
// VGPR layouts (CDNA5 ISA 7.12.2):
//   A 16x4 f32 : lanes 0-15 -> M=lane, VGPR0=K0, VGPR1=K1; lanes 16-31 -> K2,K3
//   B 4x16 f32 : mirrored (lane&15 = column N, lane half = K pair)
//   D 16x16 f32: VGPR r: lanes 0-15 -> M=r; lanes 16-31 -> M=r+8
// After the K loop: acc[0] lanes 0-15 = t, acc[1] lanes 0-15 = g; rows >= 2
// are identically zero (A rows zeroed, C starts at 0), so a full-wave
// reduction of acc[0] equals the 16-pixel t-sum.
// ---------------------------------------------------------------------------
__global__ __launch_bounds__(64) void tg_wmma_kernel(
    const float* __restrict__ x, const float* __restrict__ thw,
    const float* __restrict__ gw, float* __restrict__ g_out,
    float* __restrict__ tsum)
{
    const int lane = threadIdx.x & 31;
    const int wave = threadIdx.x >> 5;
    const int b    = blockIdx.x / BLOCKS_PER_B;
    const int pair = (blockIdx.x % BLOCKS_PER_B) * WAVES_PER_BLOCK + wave;
    const int n0   = pair * 32;              // this wave's 32 pixels

    const int m     = lane & 15;             // A row / B column
    const int khalf = lane >> 4;             // 0 or 1
    const int koff  = khalf * 2;             // K offset for this lane half

    const float* __restrict__ xb = x + (size_t)b * CL * NPIX;
    // Branchless weight fetch: lanes m>=2 load gw but are masked to 0.
    const float* __restrict__ wp = (m == 0) ? thw : gw;
    const float  msk = (m < 2) ? 1.0f : 0.0f;

    v8f acc0 = {0.f, 0.f, 0.f, 0.f, 0.f, 0.f, 0.f, 0.f};
    v8f acc1 = {0.f, 0.f, 0.f, 0.f, 0.f, 0.f, 0.f, 0.f};

#pragma unroll 4
    for (int c = 0; c < CL; c += 4) {
        const int k0 = c + koff;
        v2f a, b0, b1;
        a.x = wp[k0]     * msk;              // shared A operand (theta/g rows)
        a.y = wp[k0 + 1] * msk;
        const float* p = xb + (size_t)k0 * NPIX + n0 + m;
        b0.x = p[0];                         // row k0,   pixels n0..n0+15
        b0.y = p[NPIX];                      // row k0+1
        b1.x = p[16];                        // row k0,   pixels n0+16..n0+31
        b1.y = p[NPIX + 16];                 // row k0+1

        // Prefetch 16 channel rows ahead: lanes 0-15 cover row kp, lanes
        // 16-31 row kp+1; column spread 2*m touches every 64B segment of the
        // 128B span. locality=3 -> near-cache prefetch (fill all levels).
        int kp = c + 16; if (kp > CL - 4) kp = CL - 4;
        __builtin_prefetch(xb + (size_t)(kp + khalf) * NPIX + n0 + 2 * m, 0, 3);

        // (neg_a, A, neg_b, B, c_mod, C, reuse_a, reuse_b)
        acc0 = __builtin_amdgcn_wmma_f32_16x16x4_f32(
            false, a, false, b0, (short)0, acc0, false, false);
        acc1 = __builtin_amdgcn_wmma_f32_16x16x4_f32(
            false, a, false, b1, (short)0, acc1, false, false);
    }

    // g per pixel: D row M=1 -> element [1], lanes 0-15.
    if (lane < 16) {
        g_out[b * NPIX + n0 + lane]      = acc0[1];
        g_out[b * NPIX + n0 + 16 + lane] = acc1[1];
    }

    // t: rows 0 of both accumulators; rows seen by lanes 16-31 (M=8) are 0.
    float t = acc0[0] + acc1[0];
#pragma unroll
    for (int off = 16; off > 0; off >>= 1) t += __shfl_xor(t, off, 32);
    if (lane == 0) atomicAdd(&tsum[b], t);
}

// ---------------------------------------------------------------------------
// Pass 2: out[b,c,n] = x[b,c,n] + (tsum[b]/n * g[b,n]) * h_w[c] + h_b[c]
// One block per (b,c) row; b128 loads, non-temporal b128 stores.
// ---------------------------------------------------------------------------
__global__ __launch_bounds__(256) void epilogue_kernel(
    const float* __restrict__ x, const float* __restrict__ g,
    const float* __restrict__ tsum, const float* __restrict__ hw,
    const float* __restrict__ hb, float* __restrict__ out)
{
    const int row = blockIdx.x;              // b*CL + c
    const int b   = row >> 9;                // /512
    const int c   = row & (CL - 1);

    const float att = tsum[b] * (1.0f / (float)NPIX);
    const float hwc = hw[c] * att;           // fold mean_t into h_w[c]
    const float hbc = hb[c];

    const v4f* __restrict__ xr = (const v4f*)(x + (size_t)row * NPIX);
    const v4f* __restrict__ gr = (const v4f*)(g + (size_t)b * NPIX);
    v4f* __restrict__ orow     = (v4f*)(out + (size_t)row * NPIX);

    for (int i = threadIdx.x; i < NPIX / 4; i += 256) {
        v4f xv = xr[i];
        v4f gv = gr[i];
        v4f o;
        o.x = fmaf(gv.x, hwc, xv.x + hbc);
        o.y = fmaf(gv.y, hwc, xv.y + hbc);
        o.z = fmaf(gv.z, hwc, xv.z + hbc);
        o.w = fmaf(gv.w, hwc, xv.w + hbc);
        __builtin_nontemporal_store(o, &orow[i]);   // TH=NT: don't evict x
    }
}

extern "C" void kernel_launch(void* const* d_in, const int* in_sizes, int n_in,
                              void* d_out, int out_size, void* d_ws, size_t ws_size,
                              hipStream_t stream) {
    const float* x   = (const float*)d_in[0];  // [16,64,8,56,56]
    const float* thw = (const float*)d_in[1];  // [512]
    const float* gw  = (const float*)d_in[2];  // [512]
    const float* hw  = (const float*)d_in[3];  // [512]
    const float* hb  = (const float*)d_in[4];  // [512]
    float* out = (float*)d_out;

    // Workspace: tsum[16] at offset 0 (zeroed every call -> deterministic),
    // g[16*3136] at offset 256.
    float* tsum = (float*)d_ws;
    float* g    = (float*)((char*)d_ws + 256);

    (void)hipMemsetAsync(tsum, 0, NB * sizeof(float), stream);
    tg_wmma_kernel<<<NB * BLOCKS_PER_B, WAVES_PER_BLOCK * 32, 0, stream>>>(
        x, thw, gw, g, tsum);
    epilogue_kernel<<<NB * CL, 256, 0, stream>>>(x, g, tsum, hw, hb, out);
}